// OperationalAdmm_5274219840131
// MI455X (gfx1250) — compile-verified
//
#include <hip/hip_runtime.h>
#include <math.h>

// ---------------- problem constants ----------------
#define B_ROWS   1024
#define DIM      2048
#define BD       (B_ROWS * DIM)

#define RHO      1.0f
#define LAM      0.1f
#define ZETA     0.05f
#define TAU      0.1f
#define STEP     0.1f
#define COHESION 1.0f
#define MU       0.5f
#define DP_K     10.0f

typedef float v2f __attribute__((ext_vector_type(2)));
typedef float v8f __attribute__((ext_vector_type(8)));

// ---------------- helpers ----------------
__device__ __forceinline__ float block_reduce_256(float v, float* sm) {
    sm[threadIdx.x] = v;
    __syncthreads();
    for (int off = 128; off > 0; off >>= 1) {
        if ((int)threadIdx.x < off) sm[threadIdx.x] += sm[threadIdx.x + off];
        __syncthreads();
    }
    return sm[0];   // valid for all threads; caller must sync before reusing sm
}

// ---------------- init ----------------
__global__ __launch_bounds__(256)
void init_state_kernel(const float* __restrict__ c0,
                       float* __restrict__ c, float* __restrict__ z,
                       float* __restrict__ u,
                       float* __restrict__ pcos, float* __restrict__ pchi) {
    __shared__ float smc[256];
    __shared__ float smx[256];
    float lc = 0.f, lx = 0.f;
    int base = blockIdx.x * (256 * 8) + threadIdx.x;
#pragma unroll
    for (int t = 0; t < 8; ++t) {
        int i = base + t * 256;
        float v = c0[i];
        c[i] = v; z[i] = v; u[i] = 0.f;
        lc += cosf(v);
        lx += v * v * v;
    }
    float sc = block_reduce_256(lc, smc);
    float sx = block_reduce_256(lx, smx);
    if (threadIdx.x == 0) { pcos[blockIdx.x] = sc; pchi[blockIdx.x] = sx; }
}

// S layout (floats): [0]=prev_pas [1]=prev_score [2]=acc flag (int bits)
//                    [3]=chi_orig  ints at S+4: [0]=low [1]=done
__global__ void init_scalars_kernel(const float* __restrict__ pcos,
                                    const float* __restrict__ pchi,
                                    float* __restrict__ S) {
    __shared__ float smc[256];
    __shared__ float smx[256];
    float a = 0.f, b = 0.f;
    for (int i = threadIdx.x; i < 1024; i += 256) { a += pcos[i]; b += pchi[i]; }
    float sc = block_reduce_256(a, smc);
    float sx = block_reduce_256(b, smx);
    if (threadIdx.x == 0) {
        float pas0 = fabsf(sc / (float)BD);
        float chi0 = sx / (float)BD;
        S[0] = pas0;
        S[1] = -chi0;         // drift=0 -> exp(0)=1
        S[3] = chi0;
        int* IS = (int*)(S + 4);
        IS[0] = 0;            // low
        IS[1] = 0;            // done
        ((int*)(S + 2))[0] = 0;
    }
}

// ---------------- WMMA GEMM (LDS-staged) ----------------
// MODE 0:  out = A @ W      - A                       (A = c  -> out = r)
// MODE 1:  out = A @ W^T    - A + RHO*(c - z + u)     (A = r  -> out = g)
// Block: 256 threads = 8 waves; block tile 128(M) x 64(N); wave tile 16 x 64.
// K-slab = 32, staged cooperatively:
//   As: 128x32 A slab, row stride 40 floats (16B-aligned b128 stores,
//       reduced LDS bank clustering on fragment reads).
//   Bs: 32x64 B slab in fragment-paired layout: float2 entry
//       idx = (kb4*2 + kHalf)*64 + n  holds (B[4*kb4+kHalf][n], B[4*kb4+2+kHalf][n])
//       so one ds_load_b64 per lane yields a whole WMMA B fragment.
#define ASTRIDE 40

template <int MODE>
__global__ __launch_bounds__(256)
void wmma_gemm_kernel(const float* __restrict__ A,
                      const float* __restrict__ W,
                      const float* __restrict__ c,
                      const float* __restrict__ z,
                      const float* __restrict__ u,
                      float* __restrict__ out) {
    __shared__ __align__(16) float As[128 * ASTRIDE];   // 20.0 KB
    __shared__ __align__(16) float Bs[16 * 64 * 2];     //  8.0 KB

    const int tid   = threadIdx.x;
    const int lane  = tid & 31;
    const int wave  = tid >> 5;
    const int n0    = blockIdx.x * 64;
    const int m0    = blockIdx.y * 128;

    const int mA    = lane & 15;     // A-fragment row (within wave tile)
    const int kHalf = lane >> 4;     // 0: K={0,1}, 1: K={2,3}
    const int nB    = lane & 15;     // B-fragment col

    v8f acc[4] = {v8f{}, v8f{}, v8f{}, v8f{}};

    // staging work assignment (256 threads)
    const int aRow = tid >> 1;             // 0..127
    const int aKq  = (tid & 1) * 16;       // 0 or 16
    const int b0kk = tid >> 3;             // MODE0: W row in slab, 0..31
    const int b0n  = (tid & 7) * 4;        // MODE0: col group
    const int b1n  = tid >> 2;             // MODE1: W row (output col), 0..63
    const int b1kq = (tid & 3) * 4;        // MODE1: k group

    const float* Agr = A + (size_t)(m0 + aRow) * DIM;

    for (int kb = 0; kb < DIM; kb += 32) {
        __syncthreads();
        // ---- stage A slab (coalesced b128 loads -> b128 LDS stores) ----
        __builtin_prefetch(Agr + kb + 64, 0, 3);
#pragma unroll
        for (int q = 0; q < 16; q += 4) {
            float4 va = *(const float4*)(Agr + kb + aKq + q);
            *(float4*)&As[aRow * ASTRIDE + aKq + q] = va;
        }
        // ---- stage B slab (row-contiguous W reads in both modes) ----
        if (MODE == 0) {
#pragma unroll
            for (int h = 0; h < 2; ++h) {
                int n = b0n + h * 32;
                float4 vb = *(const float4*)(W + (size_t)(kb + b0kk) * DIM + n0 + n);
                int idx  = ((b0kk >> 2) * 2 + (b0kk & 1)) * 64 + n;
                int comp = (b0kk >> 1) & 1;
                Bs[(idx + 0) * 2 + comp] = vb.x;
                Bs[(idx + 1) * 2 + comp] = vb.y;
                Bs[(idx + 2) * 2 + comp] = vb.z;
                Bs[(idx + 3) * 2 + comp] = vb.w;
            }
        } else {
#pragma unroll
            for (int h = 0; h < 2; ++h) {
                int kk0 = b1kq + h * 16;
                float4 vb = *(const float4*)(W + (size_t)(n0 + b1n) * DIM + kb + kk0);
                float vv[4] = {vb.x, vb.y, vb.z, vb.w};
#pragma unroll
                for (int i = 0; i < 4; ++i) {
                    int kk   = kk0 + i;
                    int idx  = ((kk >> 2) * 2 + (kk & 1)) * 64 + b1n;
                    int comp = (kk >> 1) & 1;
                    Bs[idx * 2 + comp] = vv[i];
                }
            }
        }
        __syncthreads();
        // ---- compute: 8 k-steps of 4, fragments straight from LDS ----
        const float* Aw = &As[(wave * 16 + mA) * ASTRIDE];
#pragma unroll
        for (int kb4 = 0; kb4 < 8; ++kb4) {
            v2f a = *(const v2f*)(Aw + kb4 * 4 + kHalf * 2);
#pragma unroll
            for (int j = 0; j < 4; ++j) {
                v2f b = *(const v2f*)&Bs[(((kb4 * 2) + kHalf) * 64 + j * 16 + nB) * 2];
                acc[j] = __builtin_amdgcn_wmma_f32_16x16x4_f32(
                    false, a, false, b, (short)0, acc[j], false, false);
            }
        }
    }

    // C/D layout: VGPR i -> row m0+wave*16 + i + 8*(lane>>4), col n0 + 16*j + (lane&15)
    const int mOut  = m0 + wave * 16 + (lane >> 4) * 8;
    const int nBase = n0 + (lane & 15);
#pragma unroll
    for (int j = 0; j < 4; ++j) {
#pragma unroll
        for (int i = 0; i < 8; ++i) {
            size_t idx = (size_t)(mOut + i) * DIM + (nBase + j * 16);
            float v = acc[j][i] - A[idx];
            if (MODE == 1) v += RHO * (c[idx] - z[idx] + u[idx]);
            out[idx] = v;
        }
    }
}

// ---------------- fused elementwise update + partial reductions ----------------
__global__ __launch_bounds__(256)
void update_kernel(const float* __restrict__ c, const float* __restrict__ g,
                   const float* __restrict__ z, const float* __restrict__ u,
                   const float* __restrict__ love_w, const float* __restrict__ love_b,
                   float* __restrict__ cn, float* __restrict__ zn,
                   float* __restrict__ un,
                   float* __restrict__ pcos, float* __restrict__ pchi) {
    __shared__ float smc[256];
    __shared__ float smx[256];
    float lc = 0.f, lx = 0.f;
    int base = blockIdx.x * (256 * 8) + threadIdx.x;
#pragma unroll
    for (int t = 0; t < 8; ++t) {
        int i = base + t * 256;
        float ci = c[i], gi = g[i], ui = u[i];
        int col = i & (DIM - 1);
        // Mohr-Coulomb yield projection
        float limit = COHESION + MU * fabsf(gi);
        float cy = copysignf(fminf(fabsf(ci), limit), ci);
        // LoveVector + gradient step
        float cv = cy * love_w[col] + love_b[col] - STEP * gi;
        // soft threshold
        float zin = cv + ui;
        float mag = fabsf(zin) - (LAM / RHO);
        float zv  = (mag > 0.f) ? copysignf(mag, zin) : 0.f;
        float uv  = ui + (cv - zv);
        cn[i] = cv; zn[i] = zv; un[i] = uv;
        lc += cosf(cv);
        lx += cv * cv * cv;
        (void)z;
    }
    float sc = block_reduce_256(lc, smc);
    float sx = block_reduce_256(lx, smx);
    if (threadIdx.x == 0) { pcos[blockIdx.x] = sc; pchi[blockIdx.x] = sx; }
}

// ---------------- acceptance gate (1 block) ----------------
__global__ void accept_kernel(const float* __restrict__ pcos,
                              const float* __restrict__ pchi,
                              float* __restrict__ S) {
    __shared__ float smc[256];
    __shared__ float smx[256];
    float a = 0.f, b = 0.f;
    for (int i = threadIdx.x; i < 1024; i += 256) { a += pcos[i]; b += pchi[i]; }
    float sc = block_reduce_256(a, smc);
    float sx = block_reduce_256(b, smx);
    if (threadIdx.x == 0) {
        float cur_pas = fabsf(sc / (float)BD);
        float chi     = sx / (float)BD;
        float prev_pas   = S[0];
        float prev_score = S[1];
        int* IS = (int*)(S + 4);
        int low  = IS[0];
        int done = IS[1];
        float drift = fabsf(cur_pas - prev_pas);
        int violation = drift > ZETA;
        low  = (cur_pas < 0.5f) ? (low + 1) : 0;
        done = done | (low > 20);
        float cur_score = -chi * expf(-drift / ZETA);
        int stable = (cur_score - prev_score) >= -TAU;
        int acc = (!violation) && stable && (!done);
        if (acc) { S[0] = cur_pas; S[1] = cur_score; }
        IS[0] = low; IS[1] = done;
        ((int*)(S + 2))[0] = acc;
    }
}

// ---------------- conditional commit ----------------
__global__ __launch_bounds__(256)
void commit_kernel(const float* __restrict__ S,
                   const float* __restrict__ cn, const float* __restrict__ zn,
                   const float* __restrict__ un,
                   float* __restrict__ c, float* __restrict__ z,
                   float* __restrict__ u) {
    int acc = ((const int*)(S + 2))[0];
    if (!acc) return;
    int base = blockIdx.x * (256 * 4) + threadIdx.x;
#pragma unroll
    for (int t = 0; t < 4; ++t) {
        int i = base + t * 256;
        c[i] = cn[i]; z[i] = zn[i]; u[i] = un[i];
    }
}

// ---------------- final status ----------------
__global__ __launch_bounds__(256)
void final_reduce_kernel(const float* __restrict__ c, const float* __restrict__ c0,
                         float* __restrict__ psq, float* __restrict__ pchi) {
    __shared__ float sms[256];
    __shared__ float smx[256];
    float ls = 0.f, lx = 0.f;
    int base = blockIdx.x * (256 * 8) + threadIdx.x;
#pragma unroll
    for (int t = 0; t < 8; ++t) {
        int i = base + t * 256;
        float d = c[i] - c0[i];
        float v = c[i];
        ls += d * d;
        lx += v * v * v;
    }
    float ss = block_reduce_256(ls, sms);
    float sx = block_reduce_256(lx, smx);
    if (threadIdx.x == 0) { psq[blockIdx.x] = ss; pchi[blockIdx.x] = sx; }
}

__global__ void final_status_kernel(const float* __restrict__ psq,
                                    const float* __restrict__ pchi,
                                    const float* __restrict__ S,
                                    float* __restrict__ out) {
    __shared__ float sms[256];
    __shared__ float smx[256];
    float a = 0.f, b = 0.f;
    for (int i = threadIdx.x; i < 1024; i += 256) { a += psq[i]; b += pchi[i]; }
    float ss = block_reduce_256(a, sms);
    float sx = block_reduce_256(b, smx);
    if (threadIdx.x == 0) {
        float diff    = sqrtf(ss);
        float chi_cur = sx / (float)BD;
        float chi_org = S[3];
        float sgc = (float)((chi_cur > 0.f) - (chi_cur < 0.f));
        float sgo = (float)((chi_org > 0.f) - (chi_org < 0.f));
        int status = (sgc != sgo) ? 2 : ((diff > 5.f) ? 2 : ((diff > 1.f) ? 1 : 0));
        out[BD] = (float)status;
    }
}

// ---------------- Drucker-Prager per-row projection ----------------
__global__ __launch_bounds__(256)
void dp_proj_kernel(const float* __restrict__ c_phys,
                    const float* __restrict__ c0,
                    float* __restrict__ out) {
    __shared__ float trow[DIM];
    __shared__ float red[256];
    int row = blockIdx.x;
    const float* cp = c_phys + (size_t)row * DIM;
    const float* co = c0     + (size_t)row * DIM;

    float s = 0.f;
    for (int i = threadIdx.x; i < DIM; i += 256) {
        float d  = cp[i] - co[i];
        float sd = rintf(d * 10.f) * 0.1f;   // round-half-even like jnp.round
        float ti = co[i] + sd;
        trow[i] = ti;
        s += ti;
    }
    float sum = block_reduce_256(s, red);
    float m = sum / (float)DIM;
    __syncthreads();

    float s2 = 0.f;
    for (int i = threadIdx.x; i < DIM; i += 256) {
        float dv = trow[i] - m;
        s2 += dv * dv;
    }
    float nsq = block_reduce_256(s2, red);
    float n = sqrtf(nsq);
    float scale = fminf(1.0f, DP_K / (n + 1e-8f));

    for (int i = threadIdx.x; i < DIM; i += 256) {
        out[(size_t)row * DIM + i] = m + (trow[i] - m) * scale;
    }
}

// ---------------- host launch ----------------
extern "C" void kernel_launch(void* const* d_in, const int* in_sizes, int n_in,
                              void* d_out, int out_size, void* d_ws, size_t ws_size,
                              hipStream_t stream) {
    const float* c0 = (const float*)d_in[0];   // (1024, 2048)
    const float* W  = (const float*)d_in[1];   // (2048, 2048)
    const float* lw = (const float*)d_in[2];   // (2048,)
    const float* lb = (const float*)d_in[3];   // (2048,)
    float* out = (float*)d_out;

    float* ws = (float*)d_ws;
    float* c   = ws + 0 * (size_t)BD;
    float* z   = ws + 1 * (size_t)BD;
    float* u   = ws + 2 * (size_t)BD;
    float* cn  = ws + 3 * (size_t)BD;
    float* zn  = ws + 4 * (size_t)BD;
    float* un  = ws + 5 * (size_t)BD;
    float* r   = ws + 6 * (size_t)BD;
    float* g   = ws + 7 * (size_t)BD;
    float* pcos = ws + 8 * (size_t)BD;       // 1024
    float* pchi = pcos + 1024;               // 1024
    float* S    = pchi + 1024;               // 8 floats (incl. int slots)

    dim3 gGemm(DIM / 64, B_ROWS / 128);      // (32, 8)

    init_state_kernel<<<1024, 256, 0, stream>>>(c0, c, z, u, pcos, pchi);
    init_scalars_kernel<<<1, 256, 0, stream>>>(pcos, pchi, S);

    for (int it = 0; it < 20; ++it) {
        wmma_gemm_kernel<0><<<gGemm, 256, 0, stream>>>(c, W, c, z, u, r);
        wmma_gemm_kernel<1><<<gGemm, 256, 0, stream>>>(r, W, c, z, u, g);
        update_kernel<<<1024, 256, 0, stream>>>(c, g, z, u, lw, lb,
                                                cn, zn, un, pcos, pchi);
        accept_kernel<<<1, 256, 0, stream>>>(pcos, pchi, S);
        commit_kernel<<<BD / 1024, 256, 0, stream>>>(S, cn, zn, un, c, z, u);
    }

    final_reduce_kernel<<<1024, 256, 0, stream>>>(c, c0, pcos, pchi);
    final_status_kernel<<<1, 256, 0, stream>>>(pcos, pchi, S, out);
    dp_proj_kernel<<<1024, 256, 0, stream>>>(c, c0, out);

    (void)in_sizes; (void)n_in; (void)out_size; (void)ws_size;
}